// FFTCore_13288628814443
// MI455X (gfx1250) — compile-verified
//
#include <hip/hip_runtime.h>
#include <hip/hip_bf16.h>
#include <stdint.h>

// ---------------------------------------------------------------------------
// 2^24-point complex fp32 FFT, four-step decomposition 4096 x 4096.
//   Pass 1: column FFT-4096 (stride-4096 access via TDM), inter-pass twiddle,
//           TDM scatter to workspace (transposed layout).
//   Pass 2: row FFT-4096 (contiguous b128 loads), TDM scatter to output.
// Each 4096-point FFT = 3 radix-16 stages, each stage done as complex
// 16x16 matmuls on V_WMMA_F32_16X16X4_F32 (full fp32 precision).
// All LDS traffic goes through explicit addrspace(3) pointers so the
// compiler emits plain 32-bit ds_load/ds_store (no generic-pointer checks).
// ---------------------------------------------------------------------------

#define FFT_N   (1 << 24)
#define FFT_N1  4096
#define FFT_N2  4096

typedef unsigned int u32x4 __attribute__((ext_vector_type(4)));
typedef int          i32x4 __attribute__((ext_vector_type(4)));
typedef int          i32x8 __attribute__((ext_vector_type(8)));
typedef float        v2f   __attribute__((ext_vector_type(2)));
typedef float        v4f   __attribute__((ext_vector_type(4)));
typedef float        v8f   __attribute__((ext_vector_type(8)));

#if defined(__gfx1250__)

// LDS (addrspace 3) pointer types: 32-bit ds addressing, no null checks.
typedef __attribute__((address_space(3))) float lfloat;
typedef __attribute__((address_space(3))) v4f   lv4f;

static __device__ inline lfloat* as_lds(void* p) {
  // Generic LDS address = {shared aperture, 32-bit offset}; the low 32 bits
  // are exactly the ds byte address.
  return (lfloat*)(unsigned)(uintptr_t)p;
}

// --- f32 WMMA: D(16x16) = A(16x4) * B(4x16) + C ---------------------------
static __device__ inline v8f wmma4(v2f a, v2f b, v8f c) {
  // 8 args: (neg_a, A, neg_b, B, c_mod, C, reuse_a, reuse_b)
  return __builtin_amdgcn_wmma_f32_16x16x4_f32(
      false, a, false, b, (short)0, c, false, false);
}

// --- TDM descriptor for a strided column: 4096 rows x 8 bytes --------------
// data_size = 8B (complex float), tile = 1 x 4096, row stride = 4096 elems.
static __device__ inline void tdm_build(unsigned long long gbyte, unsigned ldsbyte,
                                        u32x4* g0, i32x8* g1) {
  u32x4 d0;
  d0.x = 1u;                                   // count=1, is_restore=0, no gather
  d0.y = ldsbyte;                              // LDS byte address
  d0.z = (unsigned)gbyte;                      // global_addr[31:0]
  d0.w = ((unsigned)(gbyte >> 32) & 0x01FFFFFFu) | (2u << 30); // addr[56:32] | type=2
  i32x8 d1;
  d1[0] = (int)(3u << 16);        // wg_mask=0, data_size=3 (8B), no pad/iter
  d1[1] = (int)(1u << 16);        // atomic_barrier_addr=0 | tensor_dim0 = 1
  d1[2] = (int)(4096u << 16);     // tensor_dim0 hi=0 | tensor_dim1 = 4096
  d1[3] = (int)(1u << 16);        // tensor_dim1 hi=0 | tile_dim0 = 1
  d1[4] = (int)(4096u);           // tile_dim1 = 4096 | tile_dim2 = 0
  d1[5] = (int)(4096u);           // tensor_dim0_stride[31:0] = 4096 elems
  d1[6] = 0;                      // stride hi | tensor_dim1_stride lo (unused)
  d1[7] = 0;
  *g0 = d0; *g1 = d1;
}

static __device__ inline void tdm_load_col(unsigned long long gbyte, unsigned ldsbyte) {
  u32x4 g0; i32x8 g1; tdm_build(gbyte, ldsbyte, &g0, &g1);
  i32x4 z4 = {0, 0, 0, 0};
#if __clang_major__ >= 23
  i32x8 z8 = {0, 0, 0, 0, 0, 0, 0, 0};
  __builtin_amdgcn_tensor_load_to_lds(g0, g1, z4, z4, z8, 0);
#else
  __builtin_amdgcn_tensor_load_to_lds(g0, g1, z4, z4, 0);
#endif
}

static __device__ inline void tdm_store_col(unsigned long long gbyte, unsigned ldsbyte) {
  u32x4 g0; i32x8 g1; tdm_build(gbyte, ldsbyte, &g0, &g1);
  i32x4 z4 = {0, 0, 0, 0};
#if __clang_major__ >= 23
  i32x8 z8 = {0, 0, 0, 0, 0, 0, 0, 0};
  __builtin_amdgcn_tensor_store_from_lds(g0, g1, z4, z4, z8, 0);
#else
  __builtin_amdgcn_tensor_store_from_lds(g0, g1, z4, z4, 0);
#endif
}

// --- 4096-point complex FFT in LDS (interleaved re/im), 3 radix-16 stages --
// 256 threads = 8 waves; 256 DFT-16 columns per stage; 16 columns per
// complex WMMA group; 2 groups per wave per stage.
static __device__ inline void fft4096_wmma(lfloat* lds) {
  const int tid  = threadIdx.x;
  const int lane = tid & 31;
  const int wave = tid >> 5;
  const int m    = lane & 15;     // A-row (freq) / B-column (batch) index
  const int kh   = lane >> 4;     // lane-half selects K pairs / M+8

  // DFT-16 matrix as WMMA A operand chunks: Wr, Wi, and -Wi.
  v2f wr[4], wi[4], wn[4];
#pragma unroll
  for (int kk = 0; kk < 4; ++kk) {
#pragma unroll
    for (int v = 0; v < 2; ++v) {
      int k = kk * 4 + kh * 2 + v;                 // A layout: K = 2*half + vgpr
      float s, c;
      __sincosf(-0.39269908169872414f * (float)(m * k), &s, &c); // -2*pi*m*k/16
      wr[kk][v] = c; wi[kk][v] = s; wn[kk][v] = -s;
    }
  }

#pragma unroll
  for (int st = 0; st < 3; ++st) {
    const int   Ls  = 12 - 4 * st;        // L = 4096, 256, 16
    const int   hsh = Ls - 4;             // h = L/16
    const int   h   = 1 << hsh;
    const float tws = -6.283185307179586f / (float)(1 << Ls);
#pragma unroll
    for (int cc = 0; cc < 2; ++cc) {
      const int bcol = (wave * 2 + cc) * 16 + m;   // column 0..255
      const int g    = bcol >> hsh;
      const int r    = bcol & (h - 1);
      const int base = (g << Ls) + r;

      // Gather X (B operand layout: VGPR v holds K = v + 2*half, N = lane&15)
      v2f xr[4], xi[4];
#pragma unroll
      for (int kk = 0; kk < 4; ++kk) {
#pragma unroll
        for (int v = 0; v < 2; ++v) {
          int k   = kk * 4 + kh * 2 + v;
          int idx = base + (k << hsh);
          xr[kk][v] = lds[2 * idx];
          xi[kk][v] = lds[2 * idx + 1];
        }
      }

      // Complex matmul: yr = Wr*xr + (-Wi)*xi ; yi = Wr*xi + Wi*xr
      v8f ar = {0, 0, 0, 0, 0, 0, 0, 0};
      v8f ai = {0, 0, 0, 0, 0, 0, 0, 0};
#pragma unroll
      for (int kk = 0; kk < 4; ++kk) ar = wmma4(wr[kk], xr[kk], ar);
#pragma unroll
      for (int kk = 0; kk < 4; ++kk) ar = wmma4(wn[kk], xi[kk], ar);
#pragma unroll
      for (int kk = 0; kk < 4; ++kk) ai = wmma4(wr[kk], xi[kk], ai);
#pragma unroll
      for (int kk = 0; kk < 4; ++kk) ai = wmma4(wi[kk], xr[kk], ai);

      // Intra-stage twiddle W_L^{r*j} and scatter (C/D layout: M = vv + 8*half)
#pragma unroll
      for (int vv = 0; vv < 8; ++vv) {
        int   j = vv + 8 * kh;
        float s, c;
        __sincosf(tws * (float)(r * j), &s, &c);
        float yr = ar[vv] * c - ai[vv] * s;
        float yi = ar[vv] * s + ai[vv] * c;
        int   idx = base + (j << hsh);
        lds[2 * idx]     = yr;
        lds[2 * idx + 1] = yi;
      }
    }
    __syncthreads();
  }
}
#endif // __gfx1250__

// --- Pass 1: column FFT + four-step twiddle, workgroup b = n2 --------------
__global__ __launch_bounds__(256) void fft16m_pass1(const float* __restrict__ x,
                                                    float* __restrict__ ws) {
#if defined(__gfx1250__)
  __shared__ __align__(16) float lds_s[2 * FFT_N1];
  lfloat*        lds = as_lds(lds_s);
  const unsigned b   = blockIdx.x;    // n2
  const int      tid = threadIdx.x;

  if ((tid >> 5) == 0) {              // wave 0 drives the TDM
    tdm_load_col((unsigned long long)(uintptr_t)x + (unsigned long long)b * 8ull,
                 (unsigned)(uintptr_t)lds_s);
    __builtin_amdgcn_s_wait_tensorcnt(0);
  }
  __syncthreads();

  fft4096_wmma(lds);                  // ends with a barrier

  // Inter-pass twiddle: multiply element k1 by W_N^{n2*k1}.
  // Phase (b*k1) mod 2^24 is exact in 32-bit arithmetic since 2^24 | 2^32.
#pragma unroll
  for (int e = 0; e < 16; ++e) {
    int      i = tid + 256 * e;
    unsigned p = (b * (unsigned)i) & (unsigned)(FFT_N - 1);
    float s, c;
    __sincosf(-6.283185307179586f / 16777216.0f * (float)p, &s, &c);
    float vr = lds[2 * i], vi = lds[2 * i + 1];
    lds[2 * i]     = vr * c - vi * s;
    lds[2 * i + 1] = vr * s + vi * c;
  }
  __syncthreads();

  if ((tid >> 5) == 0) {              // scatter to ws[k1*4096 + n2]
    tdm_store_col((unsigned long long)(uintptr_t)ws + (unsigned long long)b * 8ull,
                  (unsigned)(uintptr_t)lds_s);
    __builtin_amdgcn_s_wait_tensorcnt(0);
  }
#endif
}

// --- Pass 2: row FFT, workgroup b = k1 -------------------------------------
__global__ __launch_bounds__(256) void fft16m_pass2(const float* __restrict__ ws,
                                                    float* __restrict__ out) {
#if defined(__gfx1250__)
  __shared__ __align__(16) float lds_s[2 * FFT_N1];
  lfloat*        lds = as_lds(lds_s);
  const unsigned b   = blockIdx.x;    // k1
  const int      tid = threadIdx.x;

  // Contiguous coalesced b128 loads of one row (4096 complex = 32 KB)
  const v4f* __restrict__ src = (const v4f*)(ws + (size_t)b * 2 * FFT_N1);
  lv4f*                   l4  = (lv4f*)lds;
#pragma unroll
  for (int e = 0; e < 8; ++e) {
    int idx = tid + 256 * e;
    l4[idx] = src[idx];
  }
  __syncthreads();

  fft4096_wmma(lds);                  // ends with a barrier

  if ((tid >> 5) == 0) {              // scatter to out[k1 + 4096*k2]
    tdm_store_col((unsigned long long)(uintptr_t)out + (unsigned long long)b * 8ull,
                  (unsigned)(uintptr_t)lds_s);
    __builtin_amdgcn_s_wait_tensorcnt(0);
  }
#endif
}

extern "C" void kernel_launch(void* const* d_in, const int* in_sizes, int n_in,
                              void* d_out, int out_size, void* d_ws, size_t ws_size,
                              hipStream_t stream) {
  const float* x   = (const float*)d_in[0];
  float*       out = (float*)d_out;
  float*       ws  = (float*)d_ws;   // needs 2^24 * 8 B = 128 MB of scratch

  fft16m_pass1<<<dim3(FFT_N2), dim3(256), 0, stream>>>(x, ws);
  fft16m_pass2<<<dim3(FFT_N1), dim3(256), 0, stream>>>(ws, out);
}